// AttnPainterOilDensity_27041114095714
// MI455X (gfx1250) — compile-verified
//
#include <hip/hip_runtime.h>
#include <hip/hip_bf16.h>

#define TOPK   10
#define NS     256      // strokes per batch image
#define BATCH  8
#define WDIM   128
#define PIX    (WDIM * WDIM)   // 16384
#define ATH    0.1f
#define DEPTH  4        // async pipeline buffers (prefetch distance DEPTH-1)

// ---------- CDNA5 async global->LDS path (ASYNCcnt) ----------
#if defined(__gfx1250__) && __has_builtin(__builtin_amdgcn_global_load_async_to_lds_b32)
#define HAVE_ASYNC_LDS 1
#else
#define HAVE_ASYNC_LDS 0
#if defined(__gfx1250__)
#warning "gfx1250: __builtin_amdgcn_global_load_async_to_lds_b32 unavailable; using direct-load fallback"
#endif
#endif

#if HAVE_ASYNC_LDS
typedef __attribute__((address_space(1))) int gas_int;  // global (__device__ int*)
typedef __attribute__((address_space(3))) int las_int;  // LDS (__shared__ int*)

__device__ __forceinline__ void async_load_f32(const float* g, float* l) {
  __builtin_amdgcn_global_load_async_to_lds_b32((gas_int*)g, (las_int*)l, 0, 0);
}
__device__ __forceinline__ void wait_async0() {
#if __has_builtin(__builtin_amdgcn_s_wait_asynccnt)
  __builtin_amdgcn_s_wait_asynccnt(0);
#else
  asm volatile("s_wait_asynccnt 0" ::: "memory");
#endif
}
__device__ __forceinline__ void wait_async3() {
#if __has_builtin(__builtin_amdgcn_s_wait_asynccnt)
  __builtin_amdgcn_s_wait_asynccnt(3);
#else
  asm volatile("s_wait_asynccnt 3" ::: "memory");
#endif
}
#endif

__device__ __forceinline__ bool wave_any(bool p) {
#if defined(__gfx1250__) && __has_builtin(__builtin_amdgcn_ballot_w32)
  return __builtin_amdgcn_ballot_w32(p) != 0u;   // wave32 ballot
#else
  return __any(p) != 0;
#endif
}

// One thread = one output pixel. Front-to-back compositing with early exit.
__global__ __launch_bounds__(256) void AttnPainterOilDensity_render(
    const float* __restrict__ color_stroke,  // [B,NS,3,W,W]
    const float* __restrict__ alpha,         // [B,NS,1,W,W]
    const float* __restrict__ strokes,       // [B*NS,8]
    float* __restrict__ out)                 // canvas [B,3,W,W] ++ den [B,1,W,W]
{
  const int tid        = threadIdx.x;
  const int blocksPerB = PIX / 256;                       // 64
  const int b          = blockIdx.x / blocksPerB;
  const int pix        = (blockIdx.x % blocksPerB) * 256 + tid;

  // Stage per-stroke size s = w*h for this batch image (uniform across pixels).
  __shared__ float sS[NS];
  {
    const float* sp = strokes + ((size_t)b * NS + tid) * 8;
    sS[tid] = sp[2] * sp[3];
  }
  __syncthreads();

  const float* alphaB = alpha        + (size_t)b * NS * PIX      + pix;  // +n*PIX
  const float* colB   = color_stroke + (size_t)b * NS * 3 * PIX  + pix;  // +(n*3+c)*PIX

  float T = 1.0f, Ar = 0.0f, Ag = 0.0f, Ab = 0.0f, Ad = 0.0f;
  int   cnt = 0;

  // ---- Pass A: masked strokes, descending index (topmost first, front-to-back)
#if HAVE_ASYNC_LDS
  // DEPTH-deep ring of alpha tiles; each wave only touches its own lane slots,
  // so per-wave s_wait_asynccnt suffices (no barriers in the divergent loop).
  // Slot for stroke n is ((NS-1-n) & (DEPTH-1)); the DMA that would overwrite
  // slot(n) is issued only when processing stroke n-1 (targets stroke n-DEPTH),
  // so no in-flight DMA targets the slot being consumed.
  __shared__ float sA[DEPTH][256];

  async_load_f32(alphaB + (size_t)(NS - 1) * PIX, &sA[0][tid]);
  async_load_f32(alphaB + (size_t)(NS - 2) * PIX, &sA[1][tid]);
  async_load_f32(alphaB + (size_t)(NS - 3) * PIX, &sA[2][tid]);

  for (int n = NS - 1;;) {
    // prefetch alpha for stroke n-3 (clamped; redundant tail loads are harmless)
    const int m  = n - (DEPTH - 1);
    const int mc = (m < 0) ? 0 : m;
    async_load_f32(alphaB + (size_t)mc * PIX, &sA[(NS - 1 - m) & (DEPTH - 1)][tid]);
    // prefetch color channels at the same distance (taken with p~0.9)
    {
      const float* pcp = colB + (size_t)mc * 3 * PIX;
      __builtin_prefetch(pcp, 0, 0);
      __builtin_prefetch(pcp + PIX, 0, 0);
      __builtin_prefetch(pcp + 2 * PIX, 0, 0);
    }
    wait_async3();                       // in-order completion -> load(n) done

    const float a = sA[(NS - 1 - n) & (DEPTH - 1)][tid];
    if ((cnt < TOPK) && (a > ATH)) {
      const float* cp = colB + (size_t)n * 3 * PIX;
      const float ta = T * a;
      Ar += ta * cp[0];
      Ag += ta * cp[PIX];
      Ab += ta * cp[2 * PIX];
      Ad += ta * sS[n];
      T *= (1.0f - a);
      ++cnt;
    }
    --n;
    if (n < 0) break;
    if (!wave_any(cnt < TOPK)) break;    // all 32 lanes have their top-10
  }
  wait_async0();                         // drain unconsumed prefetches
#else
  for (int n = NS - 1; n >= 0; --n) {
    if (!wave_any(cnt < TOPK)) break;
    if (cnt < TOPK) {
      const float a = alphaB[(size_t)n * PIX];
      if (a > ATH) {
        const float* cp = colB + (size_t)n * 3 * PIX;
        const float ta = T * a;
        Ar += ta * cp[0];
        Ag += ta * cp[PIX];
        Ab += ta * cp[2 * PIX];
        Ad += ta * sS[n];
        T *= (1.0f - a);
        ++cnt;
      }
    }
  }
#endif

  // ---- Pass B (rare): top_k zero-slots = lowest-index unmasked strokes, ascending.
  // They composite after the masked set (front-to-back), with s = 0 (stroke_s is masked).
  if (wave_any(cnt < TOPK)) {
    for (int n2 = 0; n2 < NS; ++n2) {
      if (!wave_any(cnt < TOPK)) break;
      if (cnt < TOPK) {
        const float a = alphaB[(size_t)n2 * PIX];
        if (!(a > ATH)) {
          const float* cp = colB + (size_t)n2 * 3 * PIX;
          const float ta = T * a;
          Ar += ta * cp[0];
          Ag += ta * cp[PIX];
          Ab += ta * cp[2 * PIX];
          // Ad += ta * 0  (stroke_s masked to zero for undrawn strokes)
          T *= (1.0f - a);
          ++cnt;
        }
      }
    }
  }

  // canvas0 = 1, den0 = 1 -> add residual transmittance
  float* co = out + (size_t)b * 3 * PIX + pix;
  co[0]           = Ar + T;
  co[PIX]         = Ag + T;
  co[2 * PIX]     = Ab + T;
  out[(size_t)BATCH * 3 * PIX + (size_t)b * PIX + pix] = Ad + T;
}

extern "C" void kernel_launch(void* const* d_in, const int* in_sizes, int n_in,
                              void* d_out, int out_size, void* d_ws, size_t ws_size,
                              hipStream_t stream) {
  (void)in_sizes; (void)n_in; (void)out_size; (void)d_ws; (void)ws_size;
  const float* color_stroke = (const float*)d_in[0];
  const float* alpha        = (const float*)d_in[1];
  const float* strokes      = (const float*)d_in[2];
  float*       out          = (float*)d_out;

  dim3 grid(BATCH * PIX / 256);
  dim3 block(256);
  hipLaunchKernelGGL(AttnPainterOilDensity_render, grid, block, 0, stream,
                     color_stroke, alpha, strokes, out);
}